// DigitCaps_3332894621995
// MI455X (gfx1250) — compile-verified
//
#include <hip/hip_runtime.h>
#include <math.h>

typedef float v2f __attribute__((ext_vector_type(2)));
typedef float v8f __attribute__((ext_vector_type(8)));

// Problem constants (from reference): B=32, O=1, I=4096, V=512, D=8
#define NB   32
#define NI   4096
#define NV   512
#define ND   8
#define KTOT (NI * ND)          // 32768
#define XROW (NI * ND)          // x row stride in floats (32768)
#define WROW (NV * ND)          // W stride per i in floats (4096)
#define KCHUNK_I 64             // i's per wave  -> 4096/64 = 64 k-chunks
#define NTILES   (NV / 16)      // 32 n-tiles
#define NKCH     (NI / KCHUNK_I)

// GEMM: s[32,512] += X[32,32768] * Wm[32768,512] using V_WMMA_F32_16X16X4_F32.
// One wave (32 threads) per block. blockIdx.x = kchunk*32 + ntile.
// Each wave: one 16-wide N tile, both 16-row M tiles (batch 0-15 and 16-31),
// K range = [kc*64*8, (kc+1)*64*8). B fragments (W) are loaded once and
// shared by both M tiles, so W is streamed from HBM exactly once overall.
__global__ __launch_bounds__(32)
void caps_gemm_wmma(const float* __restrict__ x,
                    const float* __restrict__ W,
                    float* __restrict__ s)
{
    const int lane = threadIdx.x;          // 0..31
    const int nt   = blockIdx.x & (NTILES - 1);
    const int kc   = blockIdx.x >> 5;      // 0..63
    const int nn   = lane & 15;            // N / M column within tile
    const int half = lane >> 4;            // 0: K=0,1 ; 1: K=2,3 (per ISA layout)
    const int v0   = nt * 16;

    // Per-lane base pointers.
    // B tile (4x16): lane holds W[i, v0+nn, d0 + 2*half + {0,1}]
    const float* wp  = W + (long)(v0 + nn) * ND + 2 * half;
    // A tile (16x4): lane holds x[m, i, d0 + 2*half + {0,1}]
    const float* xp0 = x + (long)nn * XROW + 2 * half;           // rows 0..15
    const float* xp1 = xp0 + (long)16 * XROW;                    // rows 16..31

    v8f c0 = {};   // C tile for batches 0..15
    v8f c1 = {};   // C tile for batches 16..31

    const int ibeg = kc * KCHUNK_I;
    const int iend = ibeg + KCHUNK_I;

#pragma unroll 4
    for (int i = ibeg; i < iend; ++i) {
        const long wo = (long)i * WROW;
        const long xo = (long)i * ND;
        // Two K=4 subtiles per i: d0 = 0 and d0 = 4  (covers all 8 d's)
        v2f b0  = *(const v2f*)(wp  + wo);        // d0 = 0
        v2f b1  = *(const v2f*)(wp  + wo + 4);    // d0 = 4
        v2f a00 = *(const v2f*)(xp0 + xo);
        v2f a01 = *(const v2f*)(xp0 + xo + 4);
        v2f a10 = *(const v2f*)(xp1 + xo);
        v2f a11 = *(const v2f*)(xp1 + xo + 4);

        // Interleave the two accumulator chains for pipelining.
        c0 = __builtin_amdgcn_wmma_f32_16x16x4_f32(false, a00, false, b0,
                                                   (short)0, c0, false, false);
        c1 = __builtin_amdgcn_wmma_f32_16x16x4_f32(false, a10, false, b0,
                                                   (short)0, c1, false, false);
        c0 = __builtin_amdgcn_wmma_f32_16x16x4_f32(false, a01, false, b1,
                                                   (short)0, c0, false, false);
        c1 = __builtin_amdgcn_wmma_f32_16x16x4_f32(false, a11, false, b1,
                                                   (short)0, c1, false, false);
    }

    // Split-K reduction: hardware fp32 atomic adds into s[32][512].
    // C/D layout: VGPR r of lane L holds (M = r + 8*(L/16), N = L%16).
#pragma unroll
    for (int r = 0; r < 8; ++r) {
        const int m = r + 8 * half;
        __hip_atomic_fetch_add(&s[(long)m * NV + v0 + nn], c0[r],
                               __ATOMIC_RELAXED, __HIP_MEMORY_SCOPE_AGENT);
        __hip_atomic_fetch_add(&s[(long)(m + 16) * NV + v0 + nn], c1[r],
                               __ATOMIC_RELAXED, __HIP_MEMORY_SCOPE_AGENT);
    }
}

// squash(s) per batch, write result into both output halves (t == outputs).
__global__ __launch_bounds__(NV)
void caps_squash_out(const float* __restrict__ s, float* __restrict__ out)
{
    __shared__ float sh[NV];
    const int b = blockIdx.x;
    const int v = threadIdx.x;

    const float val = s[(long)b * NV + v];
    sh[v] = val * val;
    __syncthreads();
#pragma unroll
    for (int st = NV / 2; st >= 1; st >>= 1) {
        if (v < st) sh[v] += sh[v + st];
        __syncthreads();
    }
    const float sq = sh[0];
    const float o  = val * sq / ((1.0f + sq) * sqrtf(sq));

    out[(long)b * NV + v]            = o;   // t        [32,1,512]
    out[(long)NB * NV + b * NV + v]  = o;   // outputs  [32,1,512]
}

extern "C" void kernel_launch(void* const* d_in, const int* in_sizes, int n_in,
                              void* d_out, int out_size, void* d_ws, size_t ws_size,
                              hipStream_t stream)
{
    const float* x = (const float*)d_in[0];   // [32, 4096, 8]
    const float* W = (const float*)d_in[1];   // [1, 4096, 512, 8]
    float* out = (float*)d_out;               // [32*512] twice
    float* s   = (float*)d_ws;                // [32, 512] accumulator

    (void)in_sizes; (void)n_in; (void)out_size; (void)ws_size;

    hipMemsetAsync(s, 0, (size_t)NB * NV * sizeof(float), stream);
    caps_gemm_wmma<<<dim3(NKCH * NTILES), dim3(32), 0, stream>>>(x, W, s);
    caps_squash_out<<<dim3(NB), dim3(NV), 0, stream>>>(s, out);
}